// PDEINDLayer_67774583931011
// MI455X (gfx1250) — compile-verified
//
#include <hip/hip_runtime.h>
#include <math.h>

// ---------------------------------------------------------------------------
// Problem constants (from the reference)
// ---------------------------------------------------------------------------
#define B_    16          // BS * NIND
#define G_    256         // 16*16 grid
#define KC_   6           // multi-index count (ORDER=2)
#define D1_   16
#define NDER_ 1440        // derivative constraint rows
#define MP_   1712        // M rows (= 107*16)
#define MPA_  1728        // A-buffer rows padded to 2*16 supertile (zero rows)
#define NV_   1537        // true N = 1 + G*K
#define NP_   1552        // N padded to multiple of 16 (zero-filled)
#define NT_   107         // MP_/16 tile count
#define NTS_  54          // ceil(NT_/2) 32x32 super-tiles
#define GAMMA_C 0.5f
#define ALPHA_C 0.1f

#define ASTRIDE_ ((size_t)MPA_ * NP_)

#define OUT_U0_OFF   0                      // (16,1,256)
#define OUT_U_OFF    (B_ * G_)              // (16,1,256,6)
#define OUT_EPS_OFF  (B_ * G_ + B_ * G_ * KC_)

typedef float v2f __attribute__((ext_vector_type(2)));
typedef float v8f __attribute__((ext_vector_type(8)));

// ---------------------------------------------------------------------------
// 1) zero the dense A workspace (incl. pad rows/cols)
// ---------------------------------------------------------------------------
__global__ void k_zeroA(float* __restrict__ A) {
    size_t n = (size_t)B_ * MPA_ * NP_;
    for (size_t i = (size_t)blockIdx.x * blockDim.x + threadIdx.x; i < n;
         i += (size_t)gridDim.x * blockDim.x)
        A[i] = 0.0f;
}

// ---------------------------------------------------------------------------
// 2) scatter the sparse pattern into dense A, and build rhs = b + A*c.
//    c = gamma*e0 and A[:,0] = -1 exactly on derivative rows, so
//    rhs = [rhs_in | -gamma (x1440) | iv_rhs].
// ---------------------------------------------------------------------------
__global__ void k_build(const float* __restrict__ coeffs,
                        const float* __restrict__ rhs_in,
                        const float* __restrict__ iv_in,
                        const float* __restrict__ s0,
                        const float* __restrict__ s1,
                        float* __restrict__ A, float* __restrict__ r) {
    const int b      = blockIdx.y;
    const int stride = gridDim.x * blockDim.x;
    const int tid    = blockIdx.x * blockDim.x + threadIdx.x;
    float* Ab = A + (size_t)b * ASTRIDE_;
    float* rb = r + (size_t)b * MP_;

    // equation rows: A[g, 1+g*K+k] = coeffs[b,g,k]
    for (int e = tid; e < G_ * KC_; e += stride) {
        int g = e / KC_, k = e % KC_;
        Ab[(size_t)g * NP_ + (1 + g * KC_ + k)] = coeffs[(size_t)b * G_ * KC_ + e];
    }
    // derivative rows: +1 at (gn,klo), -1 at (g,klo), -1 at col 0, -step at (g,khi)
    for (int rr = tid; rr < NDER_; rr += stride) {
        int d = rr / 720, rem = rr % 720, p = rem / 240, idx = rem % 240;
        int g, gn, klo, khi; float s;
        if (d == 0) {                       // pairs (0,1) (1,3) (2,4)
            g = idx; gn = idx + 16;
            klo = p;
            khi = (p == 0) ? 1 : (p == 1 ? 3 : 4);
            s = s0[b * 240 + idx];
        } else {                            // pairs (0,2) (1,4) (2,5)
            int i = idx / 15, j = idx % 15;
            g = i * 16 + j; gn = g + 1;
            klo = p;
            khi = (p == 0) ? 2 : (p == 1 ? 4 : 5);
            s = s1[b * 240 + idx];
        }
        float* Ar = Ab + (size_t)(G_ + rr) * NP_;
        Ar[0]                   = -1.0f;
        Ar[1 + gn * KC_ + klo]  =  1.0f;
        Ar[1 + g  * KC_ + klo]  = -1.0f;
        Ar[1 + g  * KC_ + khi]  = -s;
    }
    // initial-value rows
    for (int j = tid; j < D1_; j += stride)
        Ab[(size_t)(G_ + NDER_ + j) * NP_ + (1 + j * KC_)] = 1.0f;
    // rhs = b + A*c
    for (int m = tid; m < MP_; m += stride) {
        float v;
        if      (m < G_)         v = rhs_in[b * G_ + m];
        else if (m < G_ + NDER_) v = -GAMMA_C;
        else                     v = iv_in[b * D1_ + (m - G_ - NDER_)];
        rb[m] = v;
    }
}

// decode lower-triangular pair index p -> (i >= j)
__device__ __forceinline__ void tri_decode(int p, int& i, int& j) {
    i = (int)((sqrtf(8.0f * (float)p + 1.0f) - 1.0f) * 0.5f);
    while ((i + 1) * (i + 2) / 2 <= p) ++i;
    while (i * (i + 1) / 2 > p) --i;
    j = p - i * (i + 1) / 2;
}

// ---------------------------------------------------------------------------
// 3) AAT = A*A^T + alpha*I, lower-triangular 32x32 super-tiles.
//    One wave per super-tile, 2x2 register blocking: 4 fp32 WMMAs per 4
//    operand loads -> 8 FLOP/byte from L2 (2x the unblocked version).
//    EXEC stays all-ones: all predicates are wave-uniform, only stores gated.
// ---------------------------------------------------------------------------
__global__ void k_syrk(const float* __restrict__ A, float* __restrict__ AAT) {
    const int b = blockIdx.y;
    int I, J; tri_decode(blockIdx.x, I, J);
    const int ti0 = 2 * I, ti1 = 2 * I + 1;   // ti1/tj1 may be 107 (pad rows)
    const int tj0 = 2 * J, tj1 = 2 * J + 1;
    const float* Ab = A   + (size_t)b * ASTRIDE_;
    float*       Cb = AAT + (size_t)b * MP_ * MP_;

    const int lane = threadIdx.x;
    const int half = lane >> 4, lr = lane & 15;
    const float* ar0 = Ab + (size_t)(ti0 * 16 + lr) * NP_ + 2 * half;
    const float* ar1 = Ab + (size_t)(ti1 * 16 + lr) * NP_ + 2 * half;
    const float* br0 = Ab + (size_t)(tj0 * 16 + lr) * NP_ + 2 * half;
    const float* br1 = Ab + (size_t)(tj1 * 16 + lr) * NP_ + 2 * half;

    v8f c00 = {}, c01 = {}, c10 = {}, c11 = {};
    for (int k0 = 0; k0 < NP_; k0 += 4) {
        v2f a0 = *(const v2f*)(ar0 + k0);
        v2f a1 = *(const v2f*)(ar1 + k0);
        v2f b0 = *(const v2f*)(br0 + k0);
        v2f b1 = *(const v2f*)(br1 + k0);
        c00 = __builtin_amdgcn_wmma_f32_16x16x4_f32(false, a0, false, b0, (short)0, c00, false, false);
        c01 = __builtin_amdgcn_wmma_f32_16x16x4_f32(false, a0, false, b1, (short)0, c01, false, false);
        c10 = __builtin_amdgcn_wmma_f32_16x16x4_f32(false, a1, false, b0, (short)0, c10, false, false);
        c11 = __builtin_amdgcn_wmma_f32_16x16x4_f32(false, a1, false, b1, (short)0, c11, false, false);
    }

    // store the valid lower-triangle tiles (+ alpha on the diagonal)
    {
        const int tis[4] = { ti0, ti0, ti1, ti1 };
        const int tjs[4] = { tj0, tj1, tj0, tj1 };
        v8f accs[4] = { c00, c01, c10, c11 };
#pragma unroll
        for (int t = 0; t < 4; ++t) {
            int ti = tis[t], tj = tjs[t];
            if (ti >= NT_ || tj >= NT_ || ti < tj) continue;   // wave-uniform
            v8f acc = accs[t];
#pragma unroll
            for (int v = 0; v < 8; ++v) {
                int m = ti * 16 + v + 8 * half;
                int n = tj * 16 + lr;
                float val = acc[v];
                if (m == n) val += ALPHA_C;
                Cb[(size_t)m * MP_ + n] = val;
            }
        }
    }
}

// ---------------------------------------------------------------------------
// 4a) Cholesky: factor 16x16 diagonal block (in-place, lower)
// ---------------------------------------------------------------------------
__global__ void k_chol_diag(float* __restrict__ AAT, int kb) {
    const int b = blockIdx.x;
    float* Cb = AAT + (size_t)b * MP_ * MP_;
    __shared__ float L[16][16];
    for (int e = threadIdx.x; e < 256; e += 32)
        L[e >> 4][e & 15] = Cb[(size_t)(kb * 16 + (e >> 4)) * MP_ + kb * 16 + (e & 15)];
    __syncthreads();
    if (threadIdx.x == 0) {
        for (int kk = 0; kk < 16; ++kk) {
            float d = sqrtf(L[kk][kk]);
            L[kk][kk] = d;
            float inv = 1.0f / d;
            for (int i = kk + 1; i < 16; ++i) L[i][kk] *= inv;
            for (int j = kk + 1; j < 16; ++j)
                for (int i = j; i < 16; ++i)
                    L[i][j] -= L[i][kk] * L[j][kk];
        }
    }
    __syncthreads();
    for (int e = threadIdx.x; e < 256; e += 32) {
        int rr = e >> 4, cc = e & 15;
        Cb[(size_t)(kb * 16 + rr) * MP_ + kb * 16 + cc] = (cc <= rr) ? L[rr][cc] : 0.0f;
    }
}

// ---------------------------------------------------------------------------
// 4b) Cholesky: panel TRSM  L21 = A21 * L11^-T   (one row per thread)
// ---------------------------------------------------------------------------
__global__ void k_chol_trsm(float* __restrict__ AAT, int kb) {
    const int b = blockIdx.y;
    float* Cb = AAT + (size_t)b * MP_ * MP_;
    __shared__ float L[16][16];
    for (int e = threadIdx.x; e < 256; e += blockDim.x)
        L[e >> 4][e & 15] = Cb[(size_t)(kb * 16 + (e >> 4)) * MP_ + kb * 16 + (e & 15)];
    __syncthreads();
    int r = (kb + 1) * 16 + blockIdx.x * blockDim.x + threadIdx.x;
    if (r >= MP_) return;
    float* Ar = Cb + (size_t)r * MP_ + kb * 16;
    float x[16];
#pragma unroll
    for (int c = 0; c < 16; ++c) {
        float t = Ar[c];
        for (int j = 0; j < c; ++j) t -= x[j] * L[c][j];
        x[c] = t / L[c][c];
    }
#pragma unroll
    for (int c = 0; c < 16; ++c) Ar[c] = x[c];
}

// ---------------------------------------------------------------------------
// 4c) Cholesky: trailing SYRK update  C -= L21_i * L21_j^T  via fp32 WMMA.
//     fp32 WMMA NEG only covers C, so negate the A operand at load time.
// ---------------------------------------------------------------------------
__global__ void k_chol_syrk(float* __restrict__ AAT, int kb) {
    const int b = blockIdx.y;
    float* Cb = AAT + (size_t)b * MP_ * MP_;
    int i, j; tri_decode(blockIdx.x, i, j);
    const int ti = kb + 1 + i, tj = kb + 1 + j;
    const int lane = threadIdx.x;
    const int half = lane >> 4, lr = lane & 15;
    const float* pi = Cb + (size_t)(ti * 16 + lr) * MP_ + kb * 16 + 2 * half;
    const float* pj = Cb + (size_t)(tj * 16 + lr) * MP_ + kb * 16 + 2 * half;

    v8f acc;
#pragma unroll
    for (int v = 0; v < 8; ++v)
        acc[v] = Cb[(size_t)(ti * 16 + v + 8 * half) * MP_ + tj * 16 + lr];
#pragma unroll
    for (int k0 = 0; k0 < 16; k0 += 4) {
        v2f a  = -(*(const v2f*)(pi + k0));   // negate product
        v2f bb =  *(const v2f*)(pj + k0);
        acc = __builtin_amdgcn_wmma_f32_16x16x4_f32(false, a, false, bb,
                                                    (short)0, acc, false, false);
    }
#pragma unroll
    for (int v = 0; v < 8; ++v)
        Cb[(size_t)(ti * 16 + v + 8 * half) * MP_ + tj * 16 + lr] = acc[v];
}

// ---------------------------------------------------------------------------
// 5) forward + backward substitution per batch (vector kept in LDS)
// ---------------------------------------------------------------------------
__global__ void k_solve(const float* __restrict__ AAT,
                        const float* __restrict__ rhs, float* __restrict__ lam) {
    const int b = blockIdx.x;
    const float* Lb = AAT + (size_t)b * MP_ * MP_;
    __shared__ float s[MP_];
    for (int i = threadIdx.x; i < MP_; i += blockDim.x) s[i] = rhs[(size_t)b * MP_ + i];
    __syncthreads();
    // L y = r
    for (int k = 0; k < MP_; ++k) {
        float yk = s[k] / Lb[(size_t)k * MP_ + k];
        __syncthreads();
        if (threadIdx.x == 0) s[k] = yk;
        for (int i = k + 1 + threadIdx.x; i < MP_; i += blockDim.x)
            s[i] -= Lb[(size_t)i * MP_ + k] * yk;
        __syncthreads();
    }
    // L^T lam = y  (reads row k of L: coalesced)
    for (int k = MP_ - 1; k >= 0; --k) {
        float xk = s[k] / Lb[(size_t)k * MP_ + k];
        __syncthreads();
        if (threadIdx.x == 0) s[k] = xk;
        for (int i = threadIdx.x; i < k; i += blockDim.x)
            s[i] -= Lb[(size_t)k * MP_ + i] * xk;
        __syncthreads();
    }
    for (int i = threadIdx.x; i < MP_; i += blockDim.x) lam[(size_t)b * MP_ + i] = s[i];
}

// ---------------------------------------------------------------------------
// 6) x = A^T lam - c ; scatter outputs (u0, u, eps)
// ---------------------------------------------------------------------------
__global__ void k_xout(const float* __restrict__ A, const float* __restrict__ lam,
                       float* __restrict__ out) {
    const int b = blockIdx.y;
    __shared__ float ls[MP_];
    for (int i = threadIdx.x; i < MP_; i += blockDim.x) ls[i] = lam[(size_t)b * MP_ + i];
    __syncthreads();
    int n = blockIdx.x * blockDim.x + threadIdx.x;
    if (n >= NV_) return;
    const float* Ab = A + (size_t)b * ASTRIDE_;
    float acc = 0.0f;
    for (int m = 0; m < MP_; ++m) acc += Ab[(size_t)m * NP_ + n] * ls[m];
    if (n == 0) {
        out[OUT_EPS_OFF + b] = acc - GAMMA_C;           // eps = x[:,0]
    } else {
        int e = n - 1;
        out[OUT_U_OFF + b * (G_ * KC_) + e] = acc;      // u
        if (e % KC_ == 0) out[OUT_U0_OFF + b * G_ + e / KC_] = acc;  // u0
    }
}

// ---------------------------------------------------------------------------
// launch: deterministic pipeline on `stream`.
// workspace: A (171.6 MB) + AAT (187.6 MB) + rhs/lam  ~ 360 MB total.
// ---------------------------------------------------------------------------
extern "C" void kernel_launch(void* const* d_in, const int* in_sizes, int n_in,
                              void* d_out, int out_size, void* d_ws, size_t ws_size,
                              hipStream_t stream) {
    const float* coeffs = (const float*)d_in[0];
    const float* rhs_in = (const float*)d_in[1];
    const float* iv_in  = (const float*)d_in[2];
    const float* s0     = (const float*)d_in[3];
    const float* s1     = (const float*)d_in[4];

    float* A   = (float*)d_ws;
    float* AAT = A   + (size_t)B_ * MPA_ * NP_;
    float* rv  = AAT + (size_t)B_ * MP_ * MP_;
    float* lam = rv  + (size_t)B_ * MP_;
    float* out = (float*)d_out;

    k_zeroA<<<1024, 256, 0, stream>>>(A);
    k_build<<<dim3(8, B_), 256, 0, stream>>>(coeffs, rhs_in, iv_in, s0, s1, A, rv);

    k_syrk<<<dim3(NTS_ * (NTS_ + 1) / 2, B_), 32, 0, stream>>>(A, AAT);

    for (int kb = 0; kb < NT_; ++kb) {
        k_chol_diag<<<B_, 32, 0, stream>>>(AAT, kb);
        int nrows = MP_ - (kb + 1) * 16;
        if (nrows > 0) {
            k_chol_trsm<<<dim3((nrows + 127) / 128, B_), 128, 0, stream>>>(AAT, kb);
            int T = NT_ - 1 - kb;
            k_chol_syrk<<<dim3(T * (T + 1) / 2, B_), 32, 0, stream>>>(AAT, kb);
        }
    }

    k_solve<<<B_, 256, 0, stream>>>(AAT, rv, lam);
    k_xout<<<dim3((NV_ + 255) / 256, B_), 256, 0, stream>>>(A, lam, out);
}